// MPS_autoencoder_77936476553444
// MI455X (gfx1250) — compile-verified
//
#include <hip/hip_runtime.h>
#include <hip/hip_bf16.h>

// ---------------------------------------------------------------------------
// MPS autoencoder decode for MI455X (gfx1250, wave32, WMMA).
//   B=128, m=32, NQ=16.  L==R (k1==k2==8) -> single chain via recursive
//   doubling; final coeffs = per-b GEMM 256x256xK1024 using
//   V_WMMA_F32_16X16X4_F32 (full fp32 matrix pipe).
//   All LDS operand tiles padded to 34 floats/row so every WMMA fragment is a
//   single aligned 8-byte ds_load_b64 into an even VGPR pair (no repack movs).
// ---------------------------------------------------------------------------

typedef float v2f __attribute__((ext_vector_type(2)));
typedef float v8f __attribute__((ext_vector_type(8)));

#define BATCH   128
#define MAT     1024            // 32*32 floats per site matrix
#define BMAT    (BATCH * MAT)   // stride between chain entries (b-major inside)
#define PAD     34              // even pad: 8B-aligned float2 frags, no bank conflicts
#define MTILE   (32 * PAD)      // padded 32x32 matrix footprint in LDS (floats)

__device__ __forceinline__ v8f wmma4(v2f a, v2f b, v8f c) {
  // D = A(16x4,f32) * B(4x16,f32) + C(16x16,f32)
  return __builtin_amdgcn_wmma_f32_16x16x4_f32(
      /*neg_a=*/false, a, /*neg_b=*/false, b,
      /*c_mod=*/(short)0, c, /*reuse_a=*/false, /*reuse_b=*/false);
}

__device__ __forceinline__ v2f lds2(const float* p) {
  return *reinterpret_cast<const v2f*>(p);    // even float index -> 8B aligned
}

// ---------------------------------------------------------------------------
// Encoder MLP: x[b,4] -> relu(1024) -> relu(32) -> e[2048] -> Ms[2][B][32][32]
// One block per batch element. Tiny cost; plain VALU.
// ---------------------------------------------------------------------------
__global__ void mlp_kernel(const float* __restrict__ x,
                           const float* __restrict__ W1, const float* __restrict__ b1,
                           const float* __restrict__ W2, const float* __restrict__ b2,
                           const float* __restrict__ W3, const float* __restrict__ b3,
                           float* __restrict__ Ms) {
  __shared__ float h1[1024];
  __shared__ float h2[32];
  __shared__ float xs[4];
  const int b   = blockIdx.x;
  const int tid = threadIdx.x;
  if (tid < 4) xs[tid] = x[b * 4 + tid];
  __syncthreads();
  const float x0 = xs[0], x1 = xs[1], x2 = xs[2], x3 = xs[3];
  for (int o = tid; o < 1024; o += 256) {
    const float* w = W1 + o * 4;
    float v = fmaf(x0, w[0], fmaf(x1, w[1], fmaf(x2, w[2], fmaf(x3, w[3], b1[o]))));
    h1[o] = fmaxf(v, 0.0f);
  }
  __syncthreads();
  if (tid < 32) {
    const float* w = W2 + tid * 1024;
    float v = b2[tid];
    for (int k = 0; k < 1024; ++k) v = fmaf(h1[k], w[k], v);
    h2[tid] = fmaxf(v, 0.0f);
  }
  __syncthreads();
  for (int o = tid; o < 2048; o += 256) {
    const float* w = W3 + o * 32;
    float v = b3[o];
#pragma unroll
    for (int k = 0; k < 32; ++k) v = fmaf(h2[k], w[k], v);
    const int s   = o >> 10;        // spin index
    const int rem = o & 1023;       // i*32+j
    Ms[(size_t)s * BMAT + (size_t)b * MAT + rem] = v;
  }
}

// ---------------------------------------------------------------------------
// Doubling step: Cout[s*S+t][b] = Cin[s][b] @ Cin[t][b]   (32x32x32 matmuls)
// One block per batch element.  Each input matrix staged TWICE in LDS:
//   row-major  (A-operand frags: contiguous K pair per lane)
//   col-major  (B-operand frags: contiguous K pair per lane)
// Each wave: full 32x32 matmul = 2x2 tiles x 8 K-steps of wmma f32 16x16x4.
// ---------------------------------------------------------------------------
template <int S>
__global__ void pairprod_kernel(const float* __restrict__ Cin,
                                float* __restrict__ Cout) {
  extern __shared__ __align__(16) float sm[];   // 2 * S * MTILE floats
  float* rm = sm;                               // row-major copies
  float* cm = sm + S * MTILE;                   // col-major copies
  const int b   = blockIdx.x;
  const int tid = threadIdx.x;
  for (int idx = tid; idx < S * 1024; idx += 256) {
    const int s = idx >> 10;
    const int e = idx & 1023;
    const int r = e >> 5, c = e & 31;
    const float v = Cin[(size_t)s * BMAT + (size_t)b * MAT + e];
    rm[s * MTILE + r * PAD + c] = v;            // [row][col]
    cm[s * MTILE + c * PAD + r] = v;            // [col][row]
  }
  __syncthreads();
  const int wave = tid >> 5;
  const int lane = tid & 31;
  const int lm   = lane & 15;
  const int half = lane >> 4;
  for (int p = wave; p < S * S; p += 8) {
    const float* X = rm + (p / S) * MTILE;      // A: X rows
    const float* Y = cm + (p % S) * MTILE;      // B: Y columns
    v8f acc[2][2] = {};
#pragma unroll
    for (int kc = 0; kc < 8; ++kc) {
      const int kk = kc * 4 + half * 2;         // per-lane K pair base (even)
      const v2f a0 = lds2(X + lm * PAD + kk);          // rows lm / lm+16
      const v2f a1 = lds2(X + (lm + 16) * PAD + kk);
      const v2f b0 = lds2(Y + lm * PAD + kk);          // cols lm / lm+16
      const v2f b1 = lds2(Y + (lm + 16) * PAD + kk);
      acc[0][0] = wmma4(a0, b0, acc[0][0]);
      acc[0][1] = wmma4(a0, b1, acc[0][1]);
      acc[1][0] = wmma4(a1, b0, acc[1][0]);
      acc[1][1] = wmma4(a1, b1, acc[1][1]);
    }
    float* O = Cout + (size_t)p * BMAT + (size_t)b * MAT;
#pragma unroll
    for (int ti = 0; ti < 2; ++ti)
#pragma unroll
      for (int tj = 0; tj < 2; ++tj)
#pragma unroll
        for (int r = 0; r < 8; ++r) {
          const int row = ti * 16 + r + half * 8;   // D layout: VGPR r -> M
          const int col = tj * 16 + lm;             // N = lane%16
          O[row * 32 + col] = acc[ti][tj][r];
        }
  }
}

// ---------------------------------------------------------------------------
// coeffs[s,t,b] = sum_ij C8[s,b,i,j] * C8[t,b,j,i]
// == per-b GEMM (256 x 256, K=1024), R operand transposed per 32x32 matrix.
// Block = (b, 128x64 output region); K staged in LDS chunks of 32.
//   Lt: [s_local][k_local] row-major pad 34   (A frags contiguous)
//   Rt: [t_local][k_local] pad 34             (B frags contiguous; the per-
//       matrix transpose is folded into staging: within a chunk the needed
//       raw elements are the stride-32 column  k_local*32 + chunk).
// 8 waves x (2x2 tiles): 4 WMMA per 4 ds_load_b64 per K-step.
// Output written directly as gs[b][s*256+t].
// ---------------------------------------------------------------------------
__global__ void trace_gemm_kernel(const float* __restrict__ C8,
                                  float* __restrict__ out) {
  __shared__ __align__(16) float Lt[128 * PAD];
  __shared__ __align__(16) float Rt[64 * PAD];
  const int b    = blockIdx.x >> 3;
  const int blk  = blockIdx.x & 7;
  const int s0   = (blk >> 2) * 128;          // 2 s-blocks of 128
  const int t0   = (blk & 3) * 64;            // 4 t-blocks of 64
  const int tid  = threadIdx.x;
  const int wave = tid >> 5;
  const int lane = tid & 31;
  const int lm   = lane & 15;
  const int half = lane >> 4;
  const int ti0  = (wave >> 1) * 2;           // tile rows ti0, ti0+1   (0..7)
  const int tj0  = (wave & 1) * 2;            // tile cols tj0, tj0+1   (0..3)
  v8f acc[2][2] = {};
  const float* base = C8 + (size_t)b * MAT;
  for (int chunk = 0; chunk < 32; ++chunk) {
    // L panel: Lt[sl][kl] = C8[s0+sl][b][chunk*32 + kl]
    for (int idx = tid; idx < 128 * 32; idx += 256) {
      const int sl = idx >> 5, kl = idx & 31;
      const float* g = base + (size_t)(s0 + sl) * BMAT + chunk * 32 + kl;
      Lt[sl * PAD + kl] = *g;
      if (chunk < 31) __builtin_prefetch(g + 32, 0, 0);   // next K chunk (L2)
    }
    // R panel (transposed): Rt[tl][kl] = C8[t0+tl][b][kl*32 + chunk]
    for (int idx = tid; idx < 64 * 32; idx += 256) {
      const int tl = idx >> 5, kl = idx & 31;
      Rt[tl * PAD + kl] = base[(size_t)(t0 + tl) * BMAT + kl * 32 + chunk];
    }
    __syncthreads();
#pragma unroll
    for (int kc = 0; kc < 8; ++kc) {
      const int kk = kc * 4 + half * 2;
      const v2f a0 = lds2(Lt + (ti0 * 16 + lm) * PAD + kk);
      const v2f a1 = lds2(Lt + ((ti0 + 1) * 16 + lm) * PAD + kk);
      const v2f b0 = lds2(Rt + (tj0 * 16 + lm) * PAD + kk);
      const v2f b1 = lds2(Rt + ((tj0 + 1) * 16 + lm) * PAD + kk);
      acc[0][0] = wmma4(a0, b0, acc[0][0]);
      acc[0][1] = wmma4(a0, b1, acc[0][1]);
      acc[1][0] = wmma4(a1, b0, acc[1][0]);
      acc[1][1] = wmma4(a1, b1, acc[1][1]);
    }
    __syncthreads();
  }
  float* O = out + (size_t)b * 65536;
#pragma unroll
  for (int di = 0; di < 2; ++di)
#pragma unroll
    for (int dj = 0; dj < 2; ++dj)
#pragma unroll
      for (int r = 0; r < 8; ++r) {
        const int row = s0 + (ti0 + di) * 16 + r + half * 8;
        const int col = t0 + (tj0 + dj) * 16 + lm;
        O[(size_t)row * 256 + col] = acc[di][dj][r];
      }
}

// ---------------------------------------------------------------------------
// Row normalization: inv[b] = rsqrt(sum_i out[b][i]^2), then scale.
// ---------------------------------------------------------------------------
__global__ void norm_reduce_kernel(const float* __restrict__ out,
                                   float* __restrict__ inv) {
  __shared__ float red[256];
  const int b = blockIdx.x, tid = threadIdx.x;
  const float* p = out + (size_t)b * 65536;
  float s = 0.f;
  for (int i = tid; i < 65536; i += 256) { const float v = p[i]; s = fmaf(v, v, s); }
  red[tid] = s;
  __syncthreads();
  for (int off = 128; off > 0; off >>= 1) {
    if (tid < off) red[tid] += red[tid + off];
    __syncthreads();
  }
  if (tid == 0) inv[b] = rsqrtf(red[0]);
}

__global__ void scale_kernel(float* __restrict__ out,
                             const float* __restrict__ inv) {
  size_t i = (size_t)blockIdx.x * blockDim.x + threadIdx.x;
  const size_t n = (size_t)BATCH * 65536;
  const size_t stride = (size_t)gridDim.x * blockDim.x;
  for (; i < n; i += stride) out[i] *= inv[i >> 16];
}

// ---------------------------------------------------------------------------
extern "C" void kernel_launch(void* const* d_in, const int* in_sizes, int n_in,
                              void* d_out, int out_size, void* d_ws, size_t ws_size,
                              hipStream_t stream) {
  const float* x  = (const float*)d_in[0];
  const float* W1 = (const float*)d_in[1];
  const float* b1 = (const float*)d_in[2];
  const float* W2 = (const float*)d_in[3];
  const float* b2 = (const float*)d_in[4];
  const float* W3 = (const float*)d_in[5];
  const float* b3 = (const float*)d_in[6];
  // d_in[7] = num_qubits (16, baked into the schedule below)
  float* out = (float*)d_out;

  float* ws  = (float*)d_ws;
  float* Ms  = ws;                       //   2 * B * 1024  (C1)
  float* C2  = Ms + (size_t)2  * BMAT;   //   4 * B * 1024
  float* C4  = C2 + (size_t)4  * BMAT;   //  16 * B * 1024
  float* C8  = C4 + (size_t)16 * BMAT;   // 256 * B * 1024  (134 MB, fits L2)
  float* inv = C8 + (size_t)256 * BMAT;  // B

  mlp_kernel<<<BATCH, 256, 0, stream>>>(x, W1, b1, W2, b2, W3, b3, Ms);

  // recursive doubling: C2 = C1@C1, C4 = C2@C2, C8 = C4@C4
  pairprod_kernel<2><<<BATCH, 256, (size_t)2 * 2  * MTILE * sizeof(float), stream>>>(Ms, C2);
  pairprod_kernel<4><<<BATCH, 256, (size_t)2 * 4  * MTILE * sizeof(float), stream>>>(C2, C4);
  pairprod_kernel<16><<<BATCH, 256, (size_t)2 * 16 * MTILE * sizeof(float), stream>>>(C4, C8);

  // coeffs + normalization
  trace_gemm_kernel<<<BATCH * 8, 256, 0, stream>>>(C8, out);
  norm_reduce_kernel<<<BATCH, 256, 0, stream>>>(out, inv);
  scale_kernel<<<4096, 256, 0, stream>>>(out, inv);
}